// PatchDecodingCNN4D_60610578481428
// MI455X (gfx1250) — compile-verified
//
#include <hip/hip_runtime.h>
#include <math.h>

typedef float v2f __attribute__((ext_vector_type(2)));
typedef float v4f __attribute__((ext_vector_type(4)));
typedef float v8f __attribute__((ext_vector_type(8)));

// ---------------------------------------------------------------------------
// Stage 1: ConvTranspose3d(1024->8, k=4, s=4) on (16,1024,4,4,4).
// stride == kernel  =>  pure GEMM:
//   M = bt*64 + coarse_voxel            (1024)
//   N = oc*64 + (kd*16+kh*4+kw)         (512)
//   K = input channel                   (1024)
// Register-blocked: each wave32 computes an M16 x N64 tile (4 accumulators
// sharing one A fragment) with V_WMMA_F32_16X16X4_F32 (full fp32 precision).
// Since nBase is a multiple of 64, each wave covers exactly one oc and the
// N-group index g is the kernel-z tap kd.
// Epilogue: +bias, tanh, scatter to h1[bt][oc][z][y][x] (16,8,16,16,16).
// ---------------------------------------------------------------------------
__global__ __launch_bounds__(128) void stage1_wmma(
    const float* __restrict__ x,
    const float* __restrict__ w1,
    const float* __restrict__ b1,
    float* __restrict__ h1)
{
  const int wave  = threadIdx.x >> 5;
  const int lane  = threadIdx.x & 31;
  const int wid   = blockIdx.x * 4 + wave;   // 512 waves total
  const int mTile = wid >> 3;                // 0..63
  const int nGrp  = wid & 7;                 // 0..7  == oc
  const int mBase = mTile << 4;
  const int nBase = nGrp << 6;
  const int half  = lane >> 4;               // K-half select (ISA 16x4 fp32 layout)
  const int l16   = lane & 15;

  // A row for this lane (tile rows never cross a bt boundary: 64 % 16 == 0)
  const int mRow = mBase + l16;
  const int bt   = mRow >> 6;
  const int vox  = mRow & 63;
  const float* aRow = x  + (size_t)bt * 65536 + vox;   // + k*64
  const float* bCol = w1 + nBase + l16;                // + k*512 (+16 per group)

  v8f acc[4] = {v8f{}, v8f{}, v8f{}, v8f{}};
  for (int k0 = 0; k0 < 1024; k0 += 4) {
    const int k = k0 + (half << 1);
    v2f a;
    a.x = aRow[(size_t)k * 64];          // VGPR0 = K0|K2 across lane halves
    a.y = aRow[(size_t)(k + 1) * 64];    // VGPR1 = K1|K3
    v2f b[4];
#pragma unroll
    for (int g = 0; g < 4; ++g) {
      b[g].x = bCol[(size_t)k * 512 + (g << 4)];
      b[g].y = bCol[(size_t)(k + 1) * 512 + (g << 4)];
    }
#pragma unroll
    for (int g = 0; g < 4; ++g)
      acc[g] = __builtin_amdgcn_wmma_f32_16x16x4_f32(
          /*neg_a=*/false, a, /*neg_b=*/false, b[g],
          /*c_mod=*/(short)0, acc[g], /*reuse_a=*/false, /*reuse_b=*/false);
  }

  // C/D layout: VGPR j -> M = j (lanes 0-15) / j+8 (lanes 16-31), N = lane&15
  const int oc   = nGrp;
  const int kh   = l16 >> 2;
  const int kw   = l16 & 3;
  const float bias = b1[oc];
#pragma unroll
  for (int g = 0; g < 4; ++g) {       // g == kd
#pragma unroll
    for (int j = 0; j < 8; ++j) {
      const int m   = mBase + j + (half << 3);
      const int bt2 = m >> 6;
      const int v   = m & 63;
      const int cz = v >> 4, cy = (v >> 2) & 3, cx = v & 3;
      const int z = cz * 4 + g, y = cy * 4 + kh, xx = cx * 4 + kw;
      h1[(size_t)bt2 * 32768 + oc * 4096 + z * 256 + y * 16 + xx] =
          tanhf(acc[g][j] + bias);
    }
  }
}

// ---------------------------------------------------------------------------
// Fused stage 2 + stage 3. Each workgroup owns an 8x8x4 tile of the final
// 32^3 output for one bt (all 4 output channels):
//   phase 1: compute the stage-2 fine region (18 x 18 x 10, 8 channels, with
//            the pad-1 halo of the stride-2 conv) into LDS from h1 (K=8 GEMV)
//            + bias + tanh.  Out-of-range fine coords are conv padding -> 0.
//   phase 2: stride-2 k=4 conv out of LDS with vectorized (b64/b128) LDS
//            reads, + b3, stored in the final (B, 4, 32,32,32, T) layout.
// This avoids ever touching the 134 MB 64^3 intermediate in HBM.
// ---------------------------------------------------------------------------
__global__ __launch_bounds__(256) void stage23_fused(
    const float* __restrict__ h1,   // (16,8,16,16,16)
    const float* __restrict__ w2,   // (8,8,64)  [ic][oc][kpos]
    const float* __restrict__ b2,   // (8)
    const float* __restrict__ w3,   // (4,8,64)  [oc][ic][kpos]
    const float* __restrict__ b3,   // (4)
    float* __restrict__ out)        // (2,4,32,32,32,8)
{
  __shared__ float s2[8][10][18][18];  // [oc2][fz][fy][fx]  ~101 KB
  __shared__ float w2s[8 * 8 * 64];    // 16 KB
  __shared__ float w3s[4 * 8 * 64];    // 8 KB (16B-aligned rows: 64 floats/row)

  const int tid = threadIdx.x;
  const int tx = blockIdx.x;          // 0..3   (x tiles of 8)
  const int ty = blockIdx.y;          // 0..3   (y tiles of 8)
  const int bt = blockIdx.z >> 3;     // 0..15
  const int tz = blockIdx.z & 7;      // 0..7   (z tiles of 4)

  for (int i = tid; i < 8 * 8 * 64; i += 256) w2s[i] = w2[i];
  for (int i = tid; i < 4 * 8 * 64; i += 256) w3s[i] = w3[i];
  __syncthreads();

  // fine-region origin (stage-3 input window start: 2*o0 - 1)
  const int fx0 = tx * 16 - 1;
  const int fy0 = ty * 16 - 1;
  const int fz0 = tz * 8 - 1;
  const float* h1b = h1 + (size_t)bt * 32768;

  float bb2[8];
#pragma unroll
  for (int o = 0; o < 8; ++o) bb2[o] = b2[o];

  // ---- phase 1: stage-2 values for the tile (+halo) ----
  for (int idx = tid; idx < 10 * 18 * 18; idx += 256) {
    const int fz = idx / (18 * 18);
    const int r  = idx % (18 * 18);
    const int fy = r / 18;
    const int fx = r % 18;
    const int gz = fz0 + fz, gy = fy0 + fy, gx = fx0 + fx;
    float v[8];
    if ((unsigned)gz < 64u && (unsigned)gy < 64u && (unsigned)gx < 64u) {
      const int cz = gz >> 2, cy = gy >> 2, cx = gx >> 2;
      const int kpos = ((gz & 3) << 4) | ((gy & 3) << 2) | (gx & 3);
      const float* hp = h1b + cz * 256 + cy * 16 + cx;
#pragma unroll
      for (int o = 0; o < 8; ++o) v[o] = bb2[o];
#pragma unroll
      for (int ic = 0; ic < 8; ++ic) {
        const float hv = hp[ic * 4096];
        const float* wp = &w2s[ic * 512 + kpos];
#pragma unroll
        for (int o = 0; o < 8; ++o) v[o] += hv * wp[o * 64];
      }
#pragma unroll
      for (int o = 0; o < 8; ++o) v[o] = tanhf(v[o]);
    } else {
#pragma unroll
      for (int o = 0; o < 8; ++o) v[o] = 0.f;  // conv pad=1 region
    }
#pragma unroll
    for (int o = 0; o < 8; ++o) s2[o][fz][fy][fx] = v[o];
  }
  __syncthreads();

  // ---- phase 2: stride-2 k=4 conv from LDS, one output voxel per thread ----
  // Window rows are 4 consecutive floats at even base index (2*lx, and the fx
  // stride 18 preserves parity) -> two aligned 8-byte LDS reads per row.
  // w3 rows are 16-byte aligned -> one b128 LDS read per output channel.
  const int lz = tid >> 6;         // 0..3
  const int ly = (tid >> 3) & 7;   // 0..7
  const int lx = tid & 7;          // 0..7
  float acc[4] = {b3[0], b3[1], b3[2], b3[3]};
  for (int ic = 0; ic < 8; ++ic) {
    for (int kd = 0; kd < 4; ++kd) {
      for (int kh = 0; kh < 4; ++kh) {
        const float* row = &s2[ic][2 * lz + kd][2 * ly + kh][2 * lx];
        const v2f p0 = *(const v2f*)row;        // kw = 0,1
        const v2f p1 = *(const v2f*)(row + 2);  // kw = 2,3
        const int kb = (kd << 4) | (kh << 2);
#pragma unroll
        for (int o = 0; o < 4; ++o) {
          const v4f wv = *(const v4f*)&w3s[(o * 8 + ic) * 64 + kb];
          acc[o] += p0.x * wv.x + p0.y * wv.y + p1.x * wv.z + p1.y * wv.w;
        }
      }
    }
  }
  const int z = tz * 4 + lz, y = ty * 8 + ly, x = tx * 8 + lx;
  const int b = bt >> 3, t = bt & 7;
#pragma unroll
  for (int o = 0; o < 4; ++o)
    out[(((size_t)(b * 4 + o) * 32768 + z * 1024 + y * 32 + x) << 3) + t] = acc[o];
}

extern "C" void kernel_launch(void* const* d_in, const int* in_sizes, int n_in,
                              void* d_out, int out_size, void* d_ws, size_t ws_size,
                              hipStream_t stream) {
  (void)in_sizes; (void)n_in; (void)out_size; (void)ws_size;
  const float* x  = (const float*)d_in[0];
  const float* w1 = (const float*)d_in[1];
  const float* b1 = (const float*)d_in[2];
  const float* w2 = (const float*)d_in[3];
  const float* b2 = (const float*)d_in[4];
  const float* w3 = (const float*)d_in[5];
  const float* b3 = (const float*)d_in[6];
  float* out = (float*)d_out;
  float* h1  = (float*)d_ws;  // 16*8*16^3 floats = 2 MB scratch

  // 512 waves (64 M-tiles x 8 oc-groups), 4 wave32 tiles per 128-thread block
  stage1_wmma<<<128, 128, 0, stream>>>(x, w1, b1, h1);
  // 4 x 4 x (16 bt * 8 z-tiles) workgroups of 256 threads
  stage23_fused<<<dim3(4, 4, 128), 256, 0, stream>>>(h1, w2, b2, w3, b3, out);
}